// HeadTargetLayer_37134287241781
// MI455X (gfx1250) — compile-verified
//
#include <hip/hip_runtime.h>
#include <stdint.h>

// Problem constants (match reference)
#define NIMG 8
#define LP   32768
#define NC   81
#define MG   128
#define BG_CLS 80
#define F_UPPER 0.4f
#define F_LOWER 0.1f

typedef float v2f __attribute__((ext_vector_type(2)));
typedef float v8f __attribute__((ext_vector_type(8)));

// ---------------- CDNA5 async global->LDS helpers ----------------
__device__ __forceinline__ unsigned lds_off(const void* p) {
  // generic pointer to LDS: low 32 bits are the LDS byte offset
  return (unsigned)(unsigned long long)p;
}
__device__ __forceinline__ void async_b32(unsigned lds, unsigned voff, const void* base) {
  asm volatile("global_load_async_to_lds_b32 %0, %1, %2 offset:0"
               :: "v"(lds), "v"(voff), "s"(base) : "memory");
}
__device__ __forceinline__ void async_b128(unsigned lds, unsigned voff, const void* base) {
  asm volatile("global_load_async_to_lds_b128 %0, %1, %2 offset:0"
               :: "v"(lds), "v"(voff), "s"(base) : "memory");
}
__device__ __forceinline__ void wait_async0() {
  asm volatile("s_wait_asynccnt 0" ::: "memory");
}

// ---------------- WMMA wave reduction ----------------
// V_WMMA_F32_16X16X4_F32 with B = ones(4x16):
//   A VGPR0 = v (lane m -> A[m][0], lane 16+m -> A[m][2]), A VGPR1 = 0
//   => D[m][n] = v[m] + v[m+16] for every n.
// D layout: VGPR r holds M=r (lanes 0-15) / M=r+8 (lanes 16-31), so each lane's
// local sum of its 8 D regs covers half the rows; shfl_xor(16) completes the sum.
__device__ __forceinline__ float wave_sum_wmma(float v) {
  v2f a; a.x = v; a.y = 0.f;
  v2f bm; bm.x = 1.f; bm.y = 1.f;
  v8f c = {};
  v8f d = __builtin_amdgcn_wmma_f32_16x16x4_f32(false, a, false, bm, (short)0, c,
                                                false, false);
  float s = d[0] + d[1] + d[2] + d[3] + d[4] + d[5] + d[6] + d[7];
  s += __shfl_xor(s, 16, 32);
  return s;  // all 32 lanes hold the wave total (fixed-order, deterministic)
}

__device__ __forceinline__ float iou_box(float ax0, float ay0, float ax1, float ay1, float areaA,
                                         float bx0, float by0, float bx1, float by1) {
  float iw = fminf(ax1, bx1) - fmaxf(ax0, bx0); iw = fmaxf(iw, 0.f);
  float ih = fminf(ay1, by1) - fmaxf(ay0, by0); ih = fmaxf(ih, 0.f);
  float inter = iw * ih;
  float areaB = (bx1 - bx0) * (by1 - by0);
  return inter / fmaxf(areaA + areaB - inter, 1e-10f);
}

// ---------------- K0: init workspace ----------------
__global__ void k0_init(unsigned long long* colmax, float* partial) {
  int t = blockIdx.x * blockDim.x + threadIdx.x;
  if (t < NIMG * MG) colmax[t] = 0ull;
  if (t < NIMG * 2) partial[t] = 0.f;
}

// ---------------- K1: per-proposal pass ----------------
// grid (LP/256, NIMG), block 256
__global__ __launch_bounds__(256) void k1_proposals(
    const float* __restrict__ rois, const float* __restrict__ scores,
    const float* __restrict__ deltas, const float* __restrict__ gt,
    float4* __restrict__ pred_out, float* __restrict__ best_out,
    int* __restrict__ jbest_out, float* __restrict__ lse_out,
    unsigned long long* __restrict__ colmax) {
  __shared__ __attribute__((aligned(16))) float sgt[MG * 4];   // 2 KB gt boxes
  __shared__ float ssc[256 * NC];                              // 81 KB score tile
  __shared__ float spr[256 * 4];                               // 4 KB pred tile

  const int tid = threadIdx.x;
  const int b = blockIdx.y;
  const int i0 = blockIdx.x * 256;
  const int i = i0 + tid;

  // Stage gt boxes (2KB) via async tensor-path loads
  const float* gtb = gt + (size_t)b * MG * 4;
  if (tid < MG) async_b128(lds_off(&sgt[tid * 4]), (unsigned)(tid * 16), gtb);

  // Stage 256x81 score rows (contiguous 82944B) coalesced into LDS
  const float* scb = scores + ((size_t)b * LP + i0) * NC;
  for (int it = 0; it < NC; ++it) {
    int q = it * 256 + tid;
    async_b32(lds_off(&ssc[q]), (unsigned)(q * 4), scb);
  }
  wait_async0();
  __syncthreads();

  // Online logsumexp + first-occurrence argmax over 81 classes
  float m = -INFINITY, s = 0.f; int cmax = 0;
  for (int c = 0; c < NC; ++c) {
    float v = ssc[tid * NC + c];
    if (v > m) { s = s * expf(m - v) + 1.f; m = v; cmax = c; }
    else       { s += expf(v - m); }
  }
  float lse = m + logf(s);

  // pred = roi + deltas[argmax]  (gather: 16B/proposal instead of 1296B)
  float4 roi = ((const float4*)rois)[(size_t)b * LP + i];
  const float* dptr = deltas + ((size_t)b * LP + i) * (4 * NC) + cmax * 4;
  float4 pr;
  pr.x = roi.x + dptr[0]; pr.y = roi.y + dptr[1];
  pr.z = roi.z + dptr[2]; pr.w = roi.w + dptr[3];

  // Row max/argmax over 128 gts
  float areaA = (pr.z - pr.x) * (pr.w - pr.y);
  float best = -1.f; int jb = 0;
  for (int j = 0; j < MG; ++j) {
    float v = iou_box(pr.x, pr.y, pr.z, pr.w, areaA,
                      sgt[j * 4], sgt[j * 4 + 1], sgt[j * 4 + 2], sgt[j * 4 + 3]);
    if (v > best) { best = v; jb = j; }
  }

  size_t gi = (size_t)b * LP + i;
  pred_out[gi] = pr;
  best_out[gi] = best;
  jbest_out[gi] = jb;
  lse_out[gi] = lse;

  // Column (per-gt) max over this block's 256 preds, then one global atomic
  spr[tid * 4 + 0] = pr.x; spr[tid * 4 + 1] = pr.y;
  spr[tid * 4 + 2] = pr.z; spr[tid * 4 + 3] = pr.w;
  __syncthreads();
  if (tid < MG) {
    float bx0 = sgt[tid * 4], by0 = sgt[tid * 4 + 1];
    float bx1 = sgt[tid * 4 + 2], by1 = sgt[tid * 4 + 3];
    unsigned long long key = 0ull;
    for (int p = 0; p < 256; ++p) {
      float ax0 = spr[p * 4], ay0 = spr[p * 4 + 1], ax1 = spr[p * 4 + 2], ay1 = spr[p * 4 + 3];
      float aA = (ax1 - ax0) * (ay1 - ay0);
      float v = iou_box(ax0, ay0, ax1, ay1, aA, bx0, by0, bx1, by1);
      // high word: iou bits (iou>=0 so bit order == value order); low word: inverted
      // index so ties pick the smallest global proposal index (jnp.argmax semantics)
      unsigned long long k = ((unsigned long long)__float_as_uint(v) << 32) |
                             (unsigned)(0x7FFFFFFF - (i0 + p));
      key = k > key ? k : key;
    }
    atomicMax(&colmax[b * MG + tid], key);
  }
}

// Block-wide deterministic float sum: per-wave WMMA sum -> 32 partials in LDS ->
// wave 0 WMMA-sums the partials.
__device__ __forceinline__ float block_sum_wmma(float v, float* sm32, int t) {
  float ws = wave_sum_wmma(v);
  if ((t & 31) == 0) sm32[t >> 5] = ws;
  __syncthreads();
  if (t < 32) {
    float s2 = wave_sum_wmma(sm32[t]);
    if (t == 0) sm32[0] = s2;
  }
  __syncthreads();
  float total = sm32[0];
  __syncthreads();   // sm32 reusable after return
  return total;
}

// ---------------- K3: per-image match + sampling + losses ----------------
// grid NIMG, block 1024
__global__ __launch_bounds__(1024) void k3_losses(
    const float* __restrict__ scores, const float* __restrict__ gt,
    const int* __restrict__ gtcls, const float4* __restrict__ pred,
    const float* __restrict__ best, const int* __restrict__ jbest,
    const float* __restrict__ lse, const unsigned long long* __restrict__ colmax,
    float* __restrict__ partial) {
  __shared__ signed char forced[LP];     // 32 KB: gt index forced onto proposal, -1 none
  __shared__ float sgt[MG * 4];
  __shared__ int scan[1024];
  __shared__ float smw[32];
  __shared__ int s_npos, s_w;

  const int b = blockIdx.x, t = threadIdx.x;
  for (int k = t; k < LP; k += 1024) forced[k] = -1;
  if (t < MG * 4) sgt[t] = gt[(size_t)b * MG * 4 + t];
  if (t == 0) { s_npos = 0; s_w = 0; }
  __syncthreads();
  if (t == 0) {
    for (int j = 0; j < MG; ++j) {   // ascending j: last write wins on duplicates
      unsigned long long key = colmax[b * MG + j];
      int fi = 0x7FFFFFFF - (int)(unsigned)(key & 0xFFFFFFFFull);
      forced[fi] = (signed char)j;
    }
  }
  __syncthreads();

  const float* bb = best + (size_t)b * LP;
  const int* jbb = jbest + (size_t)b * LP;
  const float* lsb = lse + (size_t)b * LP;
  const float4* prb = pred + (size_t)b * LP;
  const int* gcl = gtcls + (size_t)b * MG;
  const float* scb = scores + (size_t)b * LP * NC;

  // Phase 1: count positives / per-thread negatives (index-ordered chunks of 32)
  const int base = t * 32;
  int pos_cnt = 0, neg_cnt = 0;
  for (int k = 0; k < 32; ++k) {
    int ii = base + k; float bv = bb[ii];
    if (bv < F_LOWER) { neg_cnt++; continue; }
    if (forced[ii] >= 0 || bv >= F_UPPER) pos_cnt++;
  }
  atomicAdd(&s_npos, pos_cnt);
  scan[t] = neg_cnt;
  __syncthreads();
  const int bg = s_npos + 1;   // round(n_pos * 1.0) + 1

  // Inclusive Hillis-Steele scan over per-thread negative counts
  for (int d = 1; d < 1024; d <<= 1) {
    int add = (t >= d) ? scan[t - d] : 0;
    __syncthreads();
    scan[t] += add;
    __syncthreads();
  }
  int rank = scan[t] - neg_cnt;  // exclusive prefix for this thread's chunk

  // Phase 2: losses
  float ce_sum = 0.f, bb_sum = 0.f; int w_cnt = 0;
  for (int k = 0; k < 32; ++k) {
    int ii = base + k;
    float bv = bb[ii];
    bool neg = bv < F_LOWER;
    bool pos = false; int j = 0;
    if (!neg) {
      int f = forced[ii];
      if (f >= 0) { pos = true; j = f; }
      else if (bv >= F_UPPER) { pos = true; j = jbb[ii]; }
    }
    bool sel = false;
    if (neg) { rank++; sel = (rank <= bg); }
    if (pos || sel) {
      int label = pos ? gcl[j] : BG_CLS;
      ce_sum += lsb[ii] - scb[(size_t)ii * NC + label];
      w_cnt++;
    }
    if (pos) {
      float4 pr = prb[ii];
      float g0 = sgt[j * 4], g1 = sgt[j * 4 + 1], g2 = sgt[j * 4 + 2], g3 = sgt[j * 4 + 3];
      float d0 = pr.x - g0, d1 = pr.y - g1, d2 = pr.z - g2, d3 = pr.w - g3;
      float a;
      a = fabsf(d0); bb_sum += (a < 0.01f) ? 50.f * d0 * d0 : a - 0.005f;
      a = fabsf(d1); bb_sum += (a < 0.01f) ? 50.f * d1 * d1 : a - 0.005f;
      a = fabsf(d2); bb_sum += (a < 0.01f) ? 50.f * d2 * d2 : a - 0.005f;
      a = fabsf(d3); bb_sum += (a < 0.01f) ? 50.f * d3 * d3 : a - 0.005f;
    }
  }
  atomicAdd(&s_w, w_cnt);

  // Deterministic block reductions via V_WMMA_F32_16X16X4_F32
  float ce_tot = block_sum_wmma(ce_sum, smw, t);
  float bb_tot = block_sum_wmma(bb_sum, smw, t);
  if (t == 0) {
    partial[b * 2 + 0] = ce_tot / fmaxf((float)s_w, 1.f);
    partial[b * 2 + 1] = bb_tot / (float)NIMG;
  }
}

// ---------------- K4: final deterministic sum ----------------
__global__ void k4_final(const float* __restrict__ partial, float* __restrict__ out) {
  if (threadIdx.x == 0 && blockIdx.x == 0) {
    float c = 0.f, r = 0.f;
    for (int b = 0; b < NIMG; ++b) { c += partial[b * 2]; r += partial[b * 2 + 1]; }
    out[0] = c; out[1] = r;
  }
}

extern "C" void kernel_launch(void* const* d_in, const int* in_sizes, int n_in,
                              void* d_out, int out_size, void* d_ws, size_t ws_size,
                              hipStream_t stream) {
  const float* rois   = (const float*)d_in[0];
  const float* scores = (const float*)d_in[1];
  const float* deltas = (const float*)d_in[2];
  const float* gt     = (const float*)d_in[3];
  const int*   gtcls  = (const int*)d_in[4];
  float* out = (float*)d_out;

  char* w = (char*)d_ws;
  const size_t nl = (size_t)NIMG * LP;
  float4* pred = (float4*)w;                               size_t off = nl * sizeof(float4);
  float*  best = (float*)(w + off);                        off += nl * sizeof(float);
  int*    jbest = (int*)(w + off);                         off += nl * sizeof(int);
  float*  lse = (float*)(w + off);                         off += nl * sizeof(float);
  unsigned long long* colmax = (unsigned long long*)(w + off); off += (size_t)NIMG * MG * 8;
  float*  partial = (float*)(w + off);

  k0_init<<<dim3(1), dim3(1024), 0, stream>>>(colmax, partial);
  k1_proposals<<<dim3(LP / 256, NIMG), dim3(256), 0, stream>>>(
      rois, scores, deltas, gt, pred, best, jbest, lse, colmax);
  k3_losses<<<dim3(NIMG), dim3(1024), 0, stream>>>(
      scores, gt, gtcls, pred, best, jbest, lse, colmax, partial);
  k4_final<<<dim3(1), dim3(32), 0, stream>>>(partial, out);
}